// CrossAttentionHead_43739946943555
// MI455X (gfx1250) — compile-verified
//
#include <hip/hip_runtime.h>
#include <math.h>
#include <stdint.h>

typedef float v2f __attribute__((ext_vector_type(2)));
typedef float v4f __attribute__((ext_vector_type(4)));
typedef float v8f __attribute__((ext_vector_type(8)));

#define KCHUNK 32
#define LDSTR  36   // padded LDS row stride (floats): rows 16B-aligned, conflict-free b64 reads

// ---------------------------------------------------------------------------
// CDNA5 async global->LDS copy (16B per lane), tracked by ASYNCcnt.
// VDST = per-lane LDS byte address, VADDR = per-lane 64-bit global address.
// ---------------------------------------------------------------------------
__device__ __forceinline__ void async_copy_b128(unsigned lds_addr, const float* gptr) {
    asm volatile("global_load_async_to_lds_b128 %0, %1, off"
                 :: "v"(lds_addr), "v"((unsigned long long)(uintptr_t)gptr)
                 : "memory");
}
__device__ __forceinline__ void wait_asynccnt0() {
    asm volatile("s_wait_asynccnt 0x0" ::: "memory");
}

// Issue the 8 async copies (A tile 64x32 + B tile 64x32) for one K-chunk.
__device__ __forceinline__ void stage_tiles_async(const float* __restrict__ A,
                                                  const float* __restrict__ Bm,
                                                  int Kdim, int mBase, int nBase, int k0,
                                                  float* As, float* Bs,
                                                  int lrow, int slot)
{
#pragma unroll
    for (int p = 0; p < 4; ++p) {
        const int r = lrow + p * 16;
        async_copy_b128((unsigned)(uintptr_t)(As + r * LDSTR + slot * 4),
                        A + (size_t)(mBase + r) * Kdim + k0 + slot * 4);
        async_copy_b128((unsigned)(uintptr_t)(Bs + r * LDSTR + slot * 4),
                        Bm + (size_t)(nBase + r) * Kdim + k0 + slot * 4);
    }
}

// ---------------------------------------------------------------------------
// 64x64 output tile GEMM core:  C[m][n] = sum_k A[m][k] * B[n][k]
// 128 threads = 4 waves; wave w computes rows [mBase+16w, +16) x cols [nBase, +64)
// via V_WMMA_F32_16X16X4_F32. Double-buffered LDS, async global->LDS staging.
// ---------------------------------------------------------------------------
__device__ __forceinline__ void gemm64x64(const float* __restrict__ A,
                                          const float* __restrict__ Bm,
                                          int Kdim, int mBase, int nBase,
                                          float (*As)[64 * LDSTR],
                                          float (*Bs)[64 * LDSTR],
                                          v8f acc[4])
{
    const int tid  = threadIdx.x;
    const int lane = tid & 31;
    const int wave = tid >> 5;
    const int idx  = lane & 15;   // M (A frag) / N (B frag)
    const int half = lane >> 4;   // 0: K={0,1}, 1: K={2,3} within a 4-K step
    const int slot = tid & 7;     // float4 slot within a 32-float row
    const int lrow = tid >> 3;    // 16 rows per pass, 4 passes -> 64 rows

    // Prologue: stage first K-chunk
    stage_tiles_async(A, Bm, Kdim, mBase, nBase, 0, As[0], Bs[0], lrow, slot);
    wait_asynccnt0();
    __syncthreads();

    int cur = 0;
    for (int k0 = 0; k0 < Kdim; k0 += KCHUNK) {
        const int nxt = cur ^ 1;
        // Overlap: stage next chunk while computing on current buffers
        if (k0 + KCHUNK < Kdim)
            stage_tiles_async(A, Bm, Kdim, mBase, nBase, k0 + KCHUNK,
                              As[nxt], Bs[nxt], lrow, slot);

        const float* Asc = As[cur];
        const float* Bsc = Bs[cur];
#pragma unroll
        for (int kk = 0; kk < KCHUNK; kk += 4) {
            const int ko = kk + 2 * half;
            const v2f a  = *(const v2f*)(Asc + (wave * 16 + idx) * LDSTR + ko);
            const v2f b0 = *(const v2f*)(Bsc + ( 0 + idx) * LDSTR + ko);
            const v2f b1 = *(const v2f*)(Bsc + (16 + idx) * LDSTR + ko);
            const v2f b2 = *(const v2f*)(Bsc + (32 + idx) * LDSTR + ko);
            const v2f b3 = *(const v2f*)(Bsc + (48 + idx) * LDSTR + ko);
            acc[0] = __builtin_amdgcn_wmma_f32_16x16x4_f32(false, a, false, b0, (short)0, acc[0], false, false);
            acc[1] = __builtin_amdgcn_wmma_f32_16x16x4_f32(false, a, false, b1, (short)0, acc[1], false, false);
            acc[2] = __builtin_amdgcn_wmma_f32_16x16x4_f32(false, a, false, b2, (short)0, acc[2], false, false);
            acc[3] = __builtin_amdgcn_wmma_f32_16x16x4_f32(false, a, false, b3, (short)0, acc[3], false, false);
        }
        // Next buffer's async copies done + everyone finished reading current buffer
        wait_asynccnt0();
        __syncthreads();
        cur = nxt;
    }
}

// ---------------------------------------------------------------------------
// Projection + bias + interleaved RoPE:  Out[b][s][h] = rope(X[b][s][:]·W[h][:] + bias[h])
// ---------------------------------------------------------------------------
__global__ __launch_bounds__(128)
void proj_rope_kernel(const float* __restrict__ X, const float* __restrict__ W,
                      const float* __restrict__ bias, float* __restrict__ Out,
                      int S /* rows per batch */)
{
    const int Dd = 1024, Hh = 1024, ROT = 512;
    __shared__ float As[2][64 * LDSTR];
    __shared__ float Bs[2][64 * LDSTR];

    const int mBase = blockIdx.x * 64;
    const int nBase = blockIdx.y * 64;
    const int b     = blockIdx.z;

    const float* Xb = X + (size_t)b * S * Dd;
    float*       Ob = Out + (size_t)b * S * Hh;

    v8f acc[4] = {};
    gemm64x64(Xb, W, Dd, mBase, nBase, As, Bs, acc);

    const int lane = threadIdx.x & 31;
    const int wave = threadIdx.x >> 5;
    const int idx  = lane & 15;
    const int half = lane >> 4;

#pragma unroll
    for (int nt = 0; nt < 4; ++nt) {
        const int h  = nBase + nt * 16 + idx;          // output channel
        const float bv = bias[h];
        const bool doRope = (nBase + nt * 16) < ROT;   // wave-uniform (512 % 16 == 0)
        float invf = 0.0f;
        if (doRope)
            invf = __builtin_powf(10000.0f, -(float)(h & ~1) / (float)ROT);
#pragma unroll
        for (int r = 0; r < 8; ++r) {
            const int s = mBase + wave * 16 + r + 8 * half;  // sequence position
            float val = acc[nt][r] + bv;
            if (doRope) {
                const float f = (float)s * invf;
                const float c = cosf(f);
                const float sn = sinf(f);
                const float partner = __shfl_xor(val, 1, 32); // pair channel (2i <-> 2i+1)
                // even h: x1*c - x2*sn ; odd h: x2*c + x1*sn  (val is this lane's own x)
                val = (h & 1) ? fmaf(partner, sn, val * c)
                              : fmaf(-partner, sn, val * c);
            }
            Ob[(size_t)s * Hh + h] = val;
        }
    }
}

// ---------------------------------------------------------------------------
// attn[b][s][e] = sum_h Q[b][s][h] * K[b][e][h], masked with -inf
// ---------------------------------------------------------------------------
__global__ __launch_bounds__(128)
void attn_mask_kernel(const float* __restrict__ Q, const float* __restrict__ Km,
                      const int* __restrict__ mask, float* __restrict__ Out)
{
    const int SA = 4096, SE = 2048, Hh = 1024;
    __shared__ float As[2][64 * LDSTR];
    __shared__ float Bs[2][64 * LDSTR];

    const int mBase = blockIdx.x * 64;   // s tile
    const int nBase = blockIdx.y * 64;   // e tile
    const int b     = blockIdx.z;

    const float* Qb = Q + (size_t)b * SA * Hh;
    const float* Kb = Km + (size_t)b * SE * Hh;

    v8f acc[4] = {};
    gemm64x64(Qb, Kb, Hh, mBase, nBase, As, Bs, acc);

    const int lane = threadIdx.x & 31;
    const int wave = threadIdx.x >> 5;
    const int idx  = lane & 15;
    const int half = lane >> 4;
    const float NEG_INF = -__builtin_inff();

#pragma unroll
    for (int nt = 0; nt < 4; ++nt) {
        const int e = nBase + nt * 16 + idx;
        const bool keep = mask[(size_t)b * SE + e] != 0;
#pragma unroll
        for (int r = 0; r < 8; ++r) {
            const int s = mBase + wave * 16 + r + 8 * half;
            Out[((size_t)b * SA + s) * (size_t)SE + e] = keep ? acc[nt][r] : NEG_INF;
        }
    }
}

// ---------------------------------------------------------------------------
// Inputs (setup_inputs order):
//  0 x_audio [4,4096,1024] f32   1 x_event [4,2048,1024] f32
//  2 event_padding_mask [4,2048] 3 W_q [1024,1024] f32  4 b_q [1024] f32
//  5 W_k [1024,1024] f32         6 b_k [1024] f32
// Output: attn [4,4096,2048] f32
// Workspace: Q (64 MB) + K (32 MB) fp32 intermediates (fit in 192 MB L2).
// ---------------------------------------------------------------------------
extern "C" void kernel_launch(void* const* d_in, const int* in_sizes, int n_in,
                              void* d_out, int out_size, void* d_ws, size_t ws_size,
                              hipStream_t stream) {
    const float* x_audio = (const float*)d_in[0];
    const float* x_event = (const float*)d_in[1];
    const int*   mask    = (const int*)d_in[2];
    const float* W_q     = (const float*)d_in[3];
    const float* b_q     = (const float*)d_in[4];
    const float* W_k     = (const float*)d_in[5];
    const float* b_k     = (const float*)d_in[6];
    float*       out     = (float*)d_out;

    const int B = 4, SA = 4096, SE = 2048, Hh = 1024;

    float* Qws = (float*)d_ws;                        // 4*4096*1024 floats
    float* Kws = Qws + (size_t)B * SA * Hh;           // 4*2048*1024 floats

    // Q projection + RoPE
    proj_rope_kernel<<<dim3(SA / 64, Hh / 64, B), 128, 0, stream>>>(
        x_audio, W_q, b_q, Qws, SA);
    // K projection + RoPE
    proj_rope_kernel<<<dim3(SE / 64, Hh / 64, B), 128, 0, stream>>>(
        x_event, W_k, b_k, Kws, SE);
    // attention scores + padding mask
    attn_mask_kernel<<<dim3(SA / 64, SE / 64, B), 128, 0, stream>>>(
        Qws, Kws, mask, out);
}